// SOKEmbedding_76828374991717
// MI455X (gfx1250) — compile-verified
//
#include <hip/hip_runtime.h>
#include <cstdint>
#include <cstddef>

// Problem geometry (fixed by the reference).
#define NUM_SLOTS 26
#define VOCAB_PER_SLOT 100000
#define EMBED_DIM 64
#define BATCH 16384
#define ROWS (BATCH * NUM_SLOTS)      // 425,984 rows of 64 fp32 (256 B)
#define UNITS (ROWS * 16)             // 6,815,744 16-byte units
#define TPB 256                       // 8 wave32 per block
#define KDEPTH 8                      // 16B-units per thread (async depth)
#define UNITS_PER_BLOCK (TPB * KDEPTH)

static_assert(UNITS % UNITS_PER_BLOCK == 0, "grid must tile exactly");

// CDNA5 async global<->LDS data movers (ASYNCcnt-tracked).
// Signature (probe-confirmed): (v4i addrspace(1)*, v4i addrspace(3)*, imm, imm)
#if defined(__gfx1250__) &&                                                  \
    __has_builtin(__builtin_amdgcn_global_load_async_to_lds_b128) &&         \
    __has_builtin(__builtin_amdgcn_global_store_async_from_lds_b128)
#define USE_ASYNC_LDS 1
#else
#define USE_ASYNC_LDS 0
#endif

typedef int v4i __attribute__((vector_size(4 * sizeof(int))));
typedef __attribute__((address_space(1))) v4i* g_v4i_p;  // global
typedef __attribute__((address_space(3))) v4i* l_v4i_p;  // LDS

__device__ __forceinline__ void wait_asynccnt_zero() {
#if defined(__gfx1250__)
#if __has_builtin(__builtin_amdgcn_s_wait_asynccnt)
  __builtin_amdgcn_s_wait_asynccnt(0);
#else
  asm volatile("s_wait_asynccnt 0" ::: "memory");
#endif
#endif
}

__global__ __launch_bounds__(TPB) void sok_embedding_gather(
    const long long* __restrict__ ids,   // [BATCH, NUM_SLOTS] int64
    const float* __restrict__ table,     // [NUM_SLOTS*VOCAB_PER_SLOT, 64] f32
    float* __restrict__ out) {           // [BATCH, NUM_SLOTS, 64] f32
  const int tid = threadIdx.x;
  const unsigned ubase = (unsigned)blockIdx.x * UNITS_PER_BLOCK;

  // Uniform (SGPR) byte bases; all per-lane offsets are 32-bit zero-extended
  // so loads/stores lower to the saddr (GVS) form: SGPR64 base + VGPR32 offset.
  const char* idsB   = (const char*)ids;
  const char* tableB = (const char*)table;
  char* outB         = (char*)out;

#if USE_ASYNC_LDS
  __shared__ char smem[UNITS_PER_BLOCK * 16];  // 32 KB staging

  // Phase 1: KDEPTH async gathers per thread (memory -> LDS, no VGPR data).
#pragma unroll
  for (int k = 0; k < KDEPTH; ++k) {
    const unsigned u     = ubase + (unsigned)(k * TPB + tid);  // 16B unit
    const unsigned row   = u >> 4;                             // embedding row
    const unsigned piece = u & 15u;                            // 16B chunk in row
    const unsigned slot  = row % NUM_SLOTS;
    // low 32 bits of the little-endian int64 id
    const int id = *(const int*)(idsB + (size_t)(row * 8u));
    const unsigned gid   = (unsigned)id + slot * (unsigned)VOCAB_PER_SLOT;
    const unsigned goff  = (gid << 8) + (piece << 4);          // byte offset < 2^31
    __builtin_amdgcn_global_load_async_to_lds_b128(
        (g_v4i_p)(tableB + (size_t)goff),
        (l_v4i_p)&smem[(k * TPB + tid) * 16],
        /*offset=*/0, /*cpol=*/0);
  }

  wait_asynccnt_zero();  // all LDS tiles resident

  // Phase 2: stream LDS -> out with async stores (coalesced, no VGPR data).
#pragma unroll
  for (int k = 0; k < KDEPTH; ++k) {
    const unsigned u = ubase + (unsigned)(k * TPB + tid);
    __builtin_amdgcn_global_store_async_from_lds_b128(
        (g_v4i_p)(outB + (size_t)(u << 4)),
        (l_v4i_p)&smem[(k * TPB + tid) * 16],
        /*offset=*/0, /*cpol=*/0);
  }
  // S_ENDPGM performs an implicit wait-idle, covering outstanding async stores.
#else
  // Fallback: register-staged b128 gather with KDEPTH loads in flight.
  float4 v[KDEPTH];
#pragma unroll
  for (int k = 0; k < KDEPTH; ++k) {
    const unsigned u     = ubase + (unsigned)(k * TPB + tid);
    const unsigned row   = u >> 4;
    const unsigned piece = u & 15u;
    const unsigned slot  = row % NUM_SLOTS;
    const int id = *(const int*)(idsB + (size_t)(row * 8u));
    const unsigned gid  = (unsigned)id + slot * (unsigned)VOCAB_PER_SLOT;
    const unsigned goff = (gid << 8) + (piece << 4);
    v[k] = *(const float4*)(tableB + (size_t)goff);
  }
#pragma unroll
  for (int k = 0; k < KDEPTH; ++k) {
    const unsigned u = ubase + (unsigned)(k * TPB + tid);
    *(float4*)(outB + (size_t)(u << 4)) = v[k];
  }
#endif
}

extern "C" void kernel_launch(void* const* d_in, const int* in_sizes, int n_in,
                              void* d_out, int out_size, void* d_ws, size_t ws_size,
                              hipStream_t stream) {
  (void)in_sizes; (void)n_in; (void)out_size; (void)d_ws; (void)ws_size;
  const long long* ids = (const long long*)d_in[0];  // int64 ids [16384, 26]
  const float* table   = (const float*)d_in[1];      // f32 table [2.6e6, 64]
  float* out           = (float*)d_out;              // f32 out [16384, 26, 64]

  const int blocks = UNITS / UNITS_PER_BLOCK;        // 3328
  sok_embedding_gather<<<blocks, TPB, 0, stream>>>(ids, table, out);
}